// OtherEmbedding_8022998908985
// MI455X (gfx1250) — compile-verified
//
#include <hip/hip_runtime.h>

// ---------------------------------------------------------------------------
// Hetero-SAGE forward for MI455X (gfx1250, wave32).
//   Phase 1: per-relation scatter-mean accumulation (L2-resident atomics).
//   Phase 2: fused SAGE linears via V_WMMA_F32_16X16X4_F32 + bias + ReLU.
// ---------------------------------------------------------------------------

typedef __attribute__((ext_vector_type(2))) float v2f;
typedef __attribute__((ext_vector_type(8))) float v8f;

#define DFEAT 128

// ---------------- Phase 0: zero workspace ----------------------------------
__global__ void zero_ws_kernel(float* __restrict__ p, long n) {
  long i = (long)blockIdx.x * blockDim.x + threadIdx.x;
  long stride = (long)gridDim.x * blockDim.x;
  for (; i < n; i += stride) p[i] = 0.0f;
}

// ---------------- Phase 1: scatter-add edges -------------------------------
// One wave32 per edge: lane l moves float4 at feature offset l*4 (512B/edge).
__global__ void sage_scatter_kernel(const float* __restrict__ xsrc,
                                    const int* __restrict__ src,
                                    const int* __restrict__ dst,
                                    float* __restrict__ agg,
                                    float* __restrict__ cnt,
                                    int E) {
  long t = (long)blockIdx.x * blockDim.x + threadIdx.x;
  int edge = (int)(t >> 5);
  if (edge >= E) return;
  int lane = (int)(t & 31);
  int s = src[edge];   // wave-uniform load (coalesces to one request)
  int d = dst[edge];
  float4 v = ((const float4*)(xsrc + (size_t)s * DFEAT))[lane];
  float* a = agg + (size_t)d * DFEAT + lane * 4;
  atomicAdd(a + 0, v.x);
  atomicAdd(a + 1, v.y);
  atomicAdd(a + 2, v.z);
  atomicAdd(a + 3, v.w);
  if (lane == 0) atomicAdd(cnt + d, 1.0f);
}

// ---------------- Phase 2: WMMA GEMM tile helper ---------------------------
// Accumulates C(16x16) += (ascale * Arows[16 x 128]) @ W[128x128]^T tile.
// Per ISA layouts for V_WMMA_F32_16X16X4_F32:
//   A: lane L -> row (L&15), K pair at k0 + 2*(L>>4)
//   B: lane L -> col (L&15), same K pair; B(k,n) = W[colbase+n][k]
__device__ __forceinline__ v8f accum_wmma(v8f c,
                                          const float* __restrict__ Arow,
                                          float ascale,
                                          const float* __restrict__ Brow,
                                          int koff) {
#pragma unroll
  for (int k0 = 0; k0 < DFEAT; k0 += 4) {
    v2f a = *(const v2f*)(Arow + k0 + koff);
    a.x *= ascale;
    a.y *= ascale;
    v2f b = *(const v2f*)(Brow + k0 + koff);
    c = __builtin_amdgcn_wmma_f32_16x16x4_f32(false, a, false, b, (short)0, c,
                                              false, false);
  }
  return c;
}

// out = relu( mean0 @ Wl0^T + bl0 + xdst @ Wr0^T
//           [+ mean1 @ Wl1^T + bl1 + xdst @ Wr1^T] )
// One wave per 16x16 output tile; 8 col-tiles across D=128.
__global__ void sage_gemm_kernel(
    const float* __restrict__ agg0, const float* __restrict__ cnt0,
    const float* __restrict__ Wl0, const float* __restrict__ bl0,
    const float* __restrict__ Wr0,
    const float* __restrict__ agg1, const float* __restrict__ cnt1,
    const float* __restrict__ Wl1, const float* __restrict__ bl1,
    const float* __restrict__ Wr1,
    const float* __restrict__ xdst, float* __restrict__ out,
    int n_dst, int two_rel) {
  const int lane = (int)(threadIdx.x & 31);
  const int wave = (int)((blockIdx.x * blockDim.x + threadIdx.x) >> 5);
  const int row_tiles = (n_dst + 15) >> 4;
  if (wave >= row_tiles * 8) return;  // wave-uniform exit: EXEC stays all-1s

  const int mt = wave >> 3;
  const int nt = wave & 7;
  const int rowbase = mt << 4;
  const int colbase = nt << 4;
  const int m = lane & 15;
  const int koff = (lane >> 4) << 1;  // K sub-offset {0,2} per half-wave

  int row = rowbase + m;
  int rowc = row < n_dst ? row : (n_dst - 1);  // clamp; invalid rows unstored

  const float* xrow = xdst + (size_t)rowc * DFEAT;
  const int ncol = colbase + m;  // this lane's output column / W row

  v8f c = {};
  {
    float inv = 1.0f / fmaxf(cnt0[rowc], 1.0f);
    c = accum_wmma(c, agg0 + (size_t)rowc * DFEAT, inv,
                   Wl0 + (size_t)ncol * DFEAT, koff);
    c = accum_wmma(c, xrow, 1.0f, Wr0 + (size_t)ncol * DFEAT, koff);
  }
  float bias = bl0[ncol];
  if (two_rel) {
    float inv = 1.0f / fmaxf(cnt1[rowc], 1.0f);
    c = accum_wmma(c, agg1 + (size_t)rowc * DFEAT, inv,
                   Wl1 + (size_t)ncol * DFEAT, koff);
    c = accum_wmma(c, xrow, 1.0f, Wr1 + (size_t)ncol * DFEAT, koff);
    bias += bl1[ncol];
  }

  // C layout: VGPR v -> row rowbase + v + 8*(lane>>4), col colbase + m
  const int rhalf = (lane >> 4) << 3;
#pragma unroll
  for (int v = 0; v < 8; ++v) {
    int r = rowbase + v + rhalf;
    if (r < n_dst)
      out[(size_t)r * DFEAT + colbase + m] = fmaxf(c[v] + bias, 0.0f);
  }
}

// ---------------------------------------------------------------------------
extern "C" void kernel_launch(void* const* d_in, const int* in_sizes, int n_in,
                              void* d_out, int out_size, void* d_ws,
                              size_t ws_size, hipStream_t stream) {
  (void)in_sizes; (void)n_in; (void)out_size; (void)ws_size;

  const float* x_station = (const float*)d_in[1];
  const float* x_machine = (const float*)d_in[2];
  const float* x_robot   = (const float*)d_in[3];
  const float* x_job     = (const float*)d_in[0];

  const int NST = 10000, NMC = 10000, NRB = 5000, E = 500000;
  const int ndst[5] = {NST, NST, NMC, NMC, NRB};

  const int* srcp[5]; const int* dstp[5];
  const float* Wl[5]; const float* bl[5]; const float* Wr[5];
  for (int r = 0; r < 5; ++r) {
    srcp[r] = (const int*)d_in[4 + 5 * r + 0];
    dstp[r] = (const int*)d_in[4 + 5 * r + 1];
    Wl[r]   = (const float*)d_in[4 + 5 * r + 2];
    bl[r]   = (const float*)d_in[4 + 5 * r + 3];
    Wr[r]   = (const float*)d_in[4 + 5 * r + 4];
  }

  // Workspace layout: 5 agg buffers then 5 count arrays (all f32).
  float* ws = (float*)d_ws;
  float* agg[5]; float* cnt[5];
  size_t off = 0;
  for (int r = 0; r < 5; ++r) { agg[r] = ws + off; off += (size_t)ndst[r] * DFEAT; }
  for (int r = 0; r < 5; ++r) { cnt[r] = ws + off; off += (size_t)ndst[r]; }

  zero_ws_kernel<<<2048, 256, 0, stream>>>(ws, (long)off);

  // Phase 1: one wave per edge, 32 lanes x float4 = full 128-float row.
  {
    long threads = (long)E * 32;
    int blocks = (int)((threads + 255) / 256);
    for (int r = 0; r < 5; ++r)
      sage_scatter_kernel<<<blocks, 256, 0, stream>>>(x_job, srcp[r], dstp[r],
                                                      agg[r], cnt[r], E);
  }

  // Phase 2: fused mean-div + GEMMs + bias + ReLU per destination type.
  float* out_s = (float*)d_out;
  float* out_m = out_s + (size_t)NST * DFEAT;
  float* out_r = out_m + (size_t)NMC * DFEAT;

  auto launch_gemm = [&](int r0, int r1, const float* xdst, float* out, int n) {
    int row_tiles = (n + 15) / 16;
    long threads = (long)row_tiles * 8 * 32;  // one wave per 16x16 tile
    int blocks = (int)((threads + 255) / 256);
    int two = (r1 >= 0) ? 1 : 0;
    int s = (r1 >= 0) ? r1 : r0;
    sage_gemm_kernel<<<blocks, 256, 0, stream>>>(
        agg[r0], cnt[r0], Wl[r0], bl[r0], Wr[r0],
        agg[s], cnt[s], Wl[s], bl[s], Wr[s],
        xdst, out, n, two);
  };

  launch_gemm(0, 1, x_station, out_s, NST);  // cbl + li  -> h_s
  launch_gemm(2, 3, x_machine, out_m, NMC);  // nd  + eb  -> h_m
  launch_gemm(4, -1, x_robot, out_r, NRB);   // hb        -> h_r
}